// EMMA_77309411328149
// MI455X (gfx1250) — compile-verified
//
#include <hip/hip_runtime.h>
#include <hip/hip_bf16.h>
#include <math.h>

typedef float v2f __attribute__((ext_vector_type(2)));
typedef float v8f __attribute__((ext_vector_type(8)));

constexpr int kBS = 16;
constexpr int kK  = 8192;
constexpr int kH  = 768;
constexpr int kG  = 5;
constexpr int kN  = kBS + kK;        // 8208
constexpr int kMT = kN / 16;         // 513 feature tiles
constexpr int kKS = kH / 4;          // 192 K-steps of 4
constexpr int kWR = 112;             // W rows: 16 q | 80 v | 5 inv | 11 zero

// workspace offsets (floats)
constexpr int OFF_W     = 0;
constexpr int OFF_A     = kWR * kH;              // 86016, [16*5]
constexpr int OFF_SLGS  = OFF_A + kBS * kG;      // [5]
constexpr int OFF_LOGPI = OFF_SLGS + kG;         // [5]
constexpr int OFF_CNT   = OFF_LOGPI + kG;        // [5]
constexpr int OFF_SUM   = OFF_CNT + kG;          // [5]
constexpr int OFF_RMAX  = OFF_SUM + kG;          // [16]
constexpr int OFF_SIM   = 86144;                 // [16][8208]
constexpr int OFF_T     = OFF_SIM + kBS * kN;    // [8208][80]
constexpr int OFF_S2    = OFF_T + kN * kBS * kG; // [8208][5]

// output offsets (floats), tuple return order
constexpr int O_GL   = 0;                        // gmm_logits [16][8208][5]
constexpr int O_GP   = kBS * kN * kG;            // gmm_prob
constexpr int O_CTSL = 2 * kBS * kN * kG;        // cts_logits [16][8208]
constexpr int O_ACC  = O_CTSL + kBS * kN;        // acc_pred
constexpr int O_RANK = O_ACC + kBS * kN;         // new_rank_mean [5]
constexpr int O_GLBL = O_RANK + kG;              // gmm_label
constexpr int O_CLBL = O_GLBL + kBS * kN;        // cts_label
constexpr int O_CDST = O_CLBL + kBS * kN;        // cts_dist

// ---------------------------------------------------------------------------
// K1: build W = [q | (q-mu)*inv | inv | 0], A[b,g], sum(lg_sigma2), logpi,
//     zero the rank-mean reduction bins. grid = 112 blocks x 128 threads.
// ---------------------------------------------------------------------------
__global__ __launch_bounds__(128) void emma_prep(
    const float* __restrict__ q, const float* __restrict__ mu,
    const float* __restrict__ lgs, const float* __restrict__ pi,
    float* __restrict__ ws) {
  __shared__ float red[128];
  const int bid = blockIdx.x, tid = threadIdx.x;
  float* Wm = ws + OFF_W;
  if (bid < 16) {  // rows 0..15: sent_q (for sample_sim = q @ a^T)
    for (int h = tid; h < kH; h += 128) Wm[bid * kH + h] = q[bid * kH + h];
    return;
  }
  if (bid < 96) {  // rows 16..95: v[b,g,:] = (q-mu)*inv ; also A[b,g]
    const int bg = bid - 16, b = bg / kG, g = bg % kG;
    float part = 0.f;
    for (int h = tid; h < kH; h += 128) {
      float inv = expf(-lgs[g * kH + h]);
      float d = q[b * kH + h] - mu[g * kH + h];
      float v = d * inv;
      Wm[bid * kH + h] = v;
      part += d * v;  // (q-mu)^2 * inv
    }
    red[tid] = part; __syncthreads();
    for (int s = 64; s > 0; s >>= 1) {
      if (tid < s) red[tid] += red[tid + s];
      __syncthreads();
    }
    if (tid == 0) ws[OFF_A + bg] = red[0];
    return;
  }
  if (bid < 101) {  // rows 96..100: inv[g,:]; also sum lg_sigma2
    const int g = bid - 96;
    float part = 0.f;
    for (int h = tid; h < kH; h += 128) {
      float l = lgs[g * kH + h];
      Wm[bid * kH + h] = expf(-l);
      part += l;
    }
    red[tid] = part; __syncthreads();
    for (int s = 64; s > 0; s >>= 1) {
      if (tid < s) red[tid] += red[tid + s];
      __syncthreads();
    }
    if (tid == 0) ws[OFF_SLGS + g] = red[0];
    return;
  }
  // rows 101..111: zero padding for the last 16-wide n-tile
  for (int h = tid; h < kH; h += 128) Wm[bid * kH + h] = 0.f;
  if (bid == 101 && tid == 0) {
    float m = pi[0];
    for (int g = 1; g < kG; ++g) m = fmaxf(m, pi[g]);
    float s = 0.f;
    for (int g = 0; g < kG; ++g) s += expf(pi[g] - m);
    float ls = logf(s);
    for (int g = 0; g < kG; ++g) ws[OFF_LOGPI + g] = pi[g] - m - ls;
    for (int g = 0; g < kG; ++g) { ws[OFF_CNT + g] = 0.f; ws[OFF_SUM + g] = 0.f; }
  }
}

// ---------------------------------------------------------------------------
// K2: WMMA f32 16x16x4 GEMM.  OUT[n, 0..111] = a[n,:] @ W^T, where the last
//     n-tile (cols 96..111) uses the elementwise-squared A operand (a^2 @ inv).
//     One wave per 16-row feature tile; EXEC all-ones per wave (uniform exit).
// ---------------------------------------------------------------------------
__global__ __launch_bounds__(256) void emma_gemm(
    const float* __restrict__ sent_k, const float* __restrict__ queue,
    const float* __restrict__ Wm, float* __restrict__ SIM,
    float* __restrict__ Tbuf, float* __restrict__ S2buf) {
  const int wave = threadIdx.x >> 5;   // wave32
  const int lane = threadIdx.x & 31;
  const int mt = blockIdx.x * 8 + wave;
  if (mt >= kMT) return;               // wave-uniform: EXEC stays all-ones
  const int r  = lane & 15;            // matrix row (A) / col N (B,D)
  const int hk = lane >> 4;            // K-half select
  // all_feature row tile: tile 0 = sent_k, tiles 1.. = queue
  const float* arow = (mt == 0) ? (sent_k + r * kH)
                                : (queue + ((mt - 1) * 16 + r) * kH);
  v8f zero = {};
  v8f acc[7];
#pragma unroll
  for (int i = 0; i < 7; ++i) acc[i] = zero;

  for (int ks = 0; ks < kKS; ++ks) {
    const int kc = ks * 4 + hk * 2;
    v2f av = *(const v2f*)(arow + kc);   // A 16x4 f32 frag (per-lane float2)
    v2f aq = av * av;                    // squared operand for S2 tile
#pragma unroll
    for (int nt = 0; nt < 7; ++nt) {
      v2f bv = *(const v2f*)(Wm + (nt * 16 + r) * kH + kc);  // B 4x16 frag
      v2f aop = (nt == 6) ? aq : av;
      acc[nt] = __builtin_amdgcn_wmma_f32_16x16x4_f32(
          false, aop, false, bv, (short)0, acc[nt], false, false);
    }
  }

  const int m0 = mt * 16;
  // D layout: acc[nt][v] = OUT[feat = m0 + v + 8*hk][col = nt*16 + r]
  {  // cols 0..15 -> SIM[b][n] transposed store, contiguous 8 floats per lane
    float* p = SIM + r * kN + m0 + hk * 8;
#pragma unroll
    for (int v = 0; v < 8; ++v) p[v] = acc[0][v];
  }
#pragma unroll
  for (int nt = 1; nt <= 5; ++nt)  // cols 16..95 -> T[n][bg]
#pragma unroll
    for (int v = 0; v < 8; ++v)
      Tbuf[(m0 + v + hk * 8) * (kBS * kG) + (nt - 1) * 16 + r] = acc[nt][v];
  if (r < kG)                      // cols 96..100 -> S2[n][g]
#pragma unroll
    for (int v = 0; v < 8; ++v)
      S2buf[(m0 + v + hk * 8) * kG + r] = acc[6][v];
}

// ---------------------------------------------------------------------------
// K3: per-b row max of SIM (for cts_logits). grid = 16 x 256.
// ---------------------------------------------------------------------------
__global__ __launch_bounds__(256) void emma_rowmax(
    const float* __restrict__ SIM, float* __restrict__ rmax) {
  __shared__ float red[256];
  const int b = blockIdx.x, tid = threadIdx.x;
  float m = -INFINITY;
  for (int n = tid; n < kN; n += 256) m = fmaxf(m, SIM[b * kN + n]);
  red[tid] = m; __syncthreads();
  for (int s = 128; s > 0; s >>= 1) {
    if (tid < s) red[tid] = fmaxf(red[tid], red[tid + s]);
    __syncthreads();
  }
  if (tid == 0) rmax[b] = red[0];
}

// ---------------------------------------------------------------------------
// K4: per-(b,n) epilogue: GMM softmax/argmax, cts labels, masks, outputs,
//     LDS-reduced global accumulation of rank-mean counts/sums.
//     grid = 513 x 256 (exactly 16*8208 threads).
// ---------------------------------------------------------------------------
__global__ __launch_bounds__(256) void emma_epi(
    const float* __restrict__ rm, float* __restrict__ ws,
    float* __restrict__ out) {
  __shared__ float sc[kG], ssum[kG];
  const int tid = threadIdx.x;
  if (tid < kG) { sc[tid] = 0.f; ssum[tid] = 0.f; }
  __syncthreads();
  const int idx = blockIdx.x * 256 + tid;   // idx = b*kN + n
  const int b = idx / kN;
  const int n = idx - b * kN;
  const float sim = ws[OFF_SIM + idx];
  const float LG2PID = 768.0f * 1.8378770664093454f;  // H * log(2*pi)

  float logits[kG];
  float mx = -INFINITY;
#pragma unroll
  for (int g = 0; g < kG; ++g) {
    float expo = ws[OFF_A + b * kG + g]
               - 2.0f * ws[OFF_T + n * (kBS * kG) + b * kG + g]
               + ws[OFF_S2 + n * kG + g];
    float lp = -0.5f * (LG2PID + ws[OFF_SLGS + g] + expo);
    float lg = ws[OFF_LOGPI + g] + lp;
    logits[g] = lg;
    mx = fmaxf(mx, lg);
  }
  float es = 0.f, en[kG];
  int pred = 0; float best = logits[0];   // first-max, like jnp.argmax
#pragma unroll
  for (int g = 0; g < kG; ++g) {
    float ln = logits[g] - mx;
    out[O_GL + idx * kG + g] = ln;        // gmm_logits (logits_norm)
    float e = expf(ln);
    en[g] = e; es += e;
    if (logits[g] > best) { best = logits[g]; pred = g; }
  }
  const float ies = 1.f / es;
#pragma unroll
  for (int g = 0; g < kG; ++g) out[O_GP + idx * kG + g] = en[g] * ies;

  int cp = 0; float bd = fabsf(sim - rm[0]);   // first-min, like jnp.argmin
#pragma unroll
  for (int g = 1; g < kG; ++g) {
    float d = fabsf(sim - rm[g]);
    if (d < bd) { bd = d; cp = g; }
  }
  const float cdist = floorf((sim + 1.f) * 0.5f * (float)kG);
  const bool eye  = (n == b);
  const bool gold = (n == ((b + 8) & 15));     // pair_label: (b+8) mod 16, n<16

  out[O_CTSL + idx] = sim - ws[OFF_RMAX + b];
  out[O_ACC  + idx] = eye ? 0.f : sim;
  out[O_GLBL + idx] = (eye || gold) ? (float)(kG - 1) : (float)cp;
  out[O_CLBL + idx] = gold ? (float)(kG - 1) : (float)pred;
  out[O_CDST + idx] = cdist;

  // rank-mean update uses UNMASKED gmm_pred (reference semantics)
  atomicAdd(&sc[pred], 1.f);
  atomicAdd(&ssum[pred], sim);
  __syncthreads();
  if (tid < kG) {
    atomicAdd(&ws[OFF_CNT + tid], sc[tid]);
    atomicAdd(&ws[OFF_SUM + tid], ssum[tid]);
  }
}

// ---------------------------------------------------------------------------
// K5: finalize EMA rank-mean (5 floats).
// ---------------------------------------------------------------------------
__global__ void emma_fin(const float* __restrict__ rm,
                         const float* __restrict__ ws,
                         float* __restrict__ out_rank) {
  const int g = threadIdx.x;
  if (g < kG) {
    float c = ws[OFF_CNT + g];
    float s = ws[OFF_SUM + g];
    float cur = s / (c + 1e-12f);
    float upd = (c != 0.f) ? 1.f : 0.f;
    out_rank[g] = (1.f - upd * 0.1f) * rm[g] + upd * 0.1f * cur;
  }
}

extern "C" void kernel_launch(void* const* d_in, const int* in_sizes, int n_in,
                              void* d_out, int out_size, void* d_ws, size_t ws_size,
                              hipStream_t stream) {
  const float* sent_q = (const float*)d_in[0];
  const float* sent_k = (const float*)d_in[1];
  const float* queue  = (const float*)d_in[2];
  const float* mu     = (const float*)d_in[3];
  const float* lgs    = (const float*)d_in[4];
  const float* pi     = (const float*)d_in[5];
  const float* rm     = (const float*)d_in[6];
  float* out = (float*)d_out;
  float* ws  = (float*)d_ws;

  emma_prep<<<112, 128, 0, stream>>>(sent_q, mu, lgs, pi, ws);
  emma_gemm<<<(kMT + 7) / 8, 256, 0, stream>>>(sent_k, queue, ws + OFF_W,
                                               ws + OFF_SIM, ws + OFF_T,
                                               ws + OFF_S2);
  emma_rowmax<<<16, 256, 0, stream>>>(ws + OFF_SIM, ws + OFF_RMAX);
  emma_epi<<<(kBS * kN) / 256, 256, 0, stream>>>(rm, ws, out);
  emma_fin<<<1, 32, 0, stream>>>(rm, ws, out + O_RANK);

  (void)in_sizes; (void)n_in; (void)out_size; (void)ws_size;
}